// SKA_29583734735594
// MI455X (gfx1250) — compile-verified
//
#include <hip/hip_runtime.h>
#include <hip/hip_bf16.h>
#include <stdint.h>

// Problem constants (from reference): N, IC, H, W, WC, KS
#define N_   16
#define IC_  512
#define H_   64
#define W_   64
#define WC_  8
#define KS_  7
#define G_   (IC_ / WC_)        // 64 channels share one weight channel
#define R_   4                  // output rows per block
#define TPB  64                 // 2 wave32 per block
#define XROWS (R_ + KS_ - 1)    // 10 rows of x (with halo)
#define XPITCH 72               // 64 + halo, padded so rows are 16B-aligned (288B)

// ---- CDNA5 async-to-LDS (ASYNCcnt) path, guarded for toolchain differences ----
#if defined(__AMDGCN__) && __has_builtin(__builtin_amdgcn_global_load_async_to_lds_b128)
#define HAVE_ASYNC 1
#else
#define HAVE_ASYNC 0
#endif

#if HAVE_ASYNC
typedef __attribute__((ext_vector_type(4))) int v4i;
typedef __attribute__((address_space(1))) v4i* gptr_v4i;   // global (AS1)
typedef __attribute__((address_space(3))) v4i* lptr_v4i;   // LDS (AS3)
#endif

__device__ __forceinline__ void async_copy16(const float* gsrc, float* ldst) {
#if HAVE_ASYNC
  // (global src, lds dst, imm offset, imm cpol)
  __builtin_amdgcn_global_load_async_to_lds_b128(
      (gptr_v4i)(uint64_t)(uintptr_t)gsrc,
      (lptr_v4i)(uint32_t)(uintptr_t)ldst,   // low 32 bits of generic = LDS offset
      0, 0);
#else
  *(float4*)ldst = *(const float4*)gsrc;   // fallback: vmem load + ds store
#endif
}

#if defined(__AMDGCN__) && __has_builtin(__builtin_amdgcn_s_wait_asynccnt)
#define WAIT_ASYNC(n) __builtin_amdgcn_s_wait_asynccnt(n)
#elif HAVE_ASYNC
#define WAIT_ASYNC(n) asm volatile("s_wait_asynccnt " #n ::: "memory")
#else
#define WAIT_ASYNC(n) ((void)0)
#endif

// Stage one channel's x tile (rows h0-3 .. h0+6, full width, zero halo) into LDS.
// Every lane issues exactly 3 async b128 ops (invalid lanes -> trash slot, clamped
// global address) so the wave's ASYNCcnt delta is a compile-time constant of 3.
__device__ __forceinline__ void stage_x(const float* __restrict__ xch,
                                        float* __restrict__ xbuf,
                                        float* __restrict__ trash,
                                        int h0, int rr, int c4) {
#pragma unroll
  for (int rb = 0; rb < 3; ++rb) {
    int row = rb * 4 + rr;            // 0..11 (rows 10,11 masked)
    int hr  = h0 - 3 + row;           // global row, may be out of image
    bool valid = (row < XROWS) && (hr >= 0) && (hr < H_);
    int hrc = hr < 0 ? 0 : (hr > (H_ - 1) ? (H_ - 1) : hr);
    const float* src = xch + (size_t)hrc * W_ + c4;
    float* dst = valid ? (xbuf + row * XPITCH + 4 + c4) : trash;
    async_copy16(src, dst);
  }
}

__global__ __launch_bounds__(TPB)
void SKA_29583734735594_kernel(const float* __restrict__ x,
                               const float* __restrict__ w,
                               float* __restrict__ out) {
  __shared__ __align__(16) float s_w[49 * R_ * W_];           // 50,176 B weight tile
  __shared__ __align__(16) float s_x[2][XROWS * XPITCH];      //  5,760 B double-buffered x
  __shared__ __align__(16) float s_trash[TPB * 4];            //  1,024 B redirect target

  const int tid = threadIdx.x;
  const int h0  = blockIdx.x * R_;     // row tile
  const int cw  = blockIdx.y;          // weight channel 0..7
  const int n   = blockIdx.z;          // batch

  const int c4 = (tid & 15) * 4;       // column group start (0,4,...,60)
  const int rr = tid >> 4;             // row within tile (0..3)
  float* mytrash = &s_trash[tid * 4];

  const size_t HW = (size_t)H_ * W_;

  // --- pre-zero both x buffers (halo columns / out-of-image rows stay zero) ---
  {
    float4 z; z.x = z.y = z.z = z.w = 0.f;
    float4* p = (float4*)&s_x[0][0];
    for (int i = tid; i < (2 * XROWS * XPITCH) / 4; i += TPB) p[i] = z;
  }
  __syncthreads();  // zeros committed to LDS before async engine writes interior

  // --- stage the whole weight tile once (reused by all 64 channels) ---
  const float* wbase = w + ((size_t)n * WC_ + cw) * 49 * HW;
#pragma unroll 1
  for (int k = 0; k < 49; ++k) {
    async_copy16(wbase + (size_t)k * HW + (size_t)(h0 + rr) * W_ + c4,
                 &s_w[k * (R_ * W_) + rr * W_ + c4]);
  }

  // --- prefetch channel 0's x tile ---
  const float* xb0 = x + ((size_t)n * IC_ + cw) * HW;   // channel c = ch*WC_ + cw
  stage_x(xb0, &s_x[0][0], mytrash, h0, rr, c4);

  for (int ch = 0; ch < G_; ++ch) {
    // Prefetch next channel into the other buffer, then retire current tile.
    if (ch + 1 < G_) {
      stage_x(xb0 + (size_t)(ch + 1) * WC_ * HW, &s_x[(ch + 1) & 1][0],
              mytrash, h0, rr, c4);
      WAIT_ASYNC(3);          // in-order completion: weights + tile(ch) done
    } else {
      WAIT_ASYNC(0);          // drain everything for the last tile
    }
    __syncthreads();          // all waves' LDS data visible

    // --- compute 4 pixels (one row, cols c4..c4+3) for this channel ---
    const float* xb = &s_x[ch & 1][0];
    float a0 = 0.f, a1 = 0.f, a2 = 0.f, a3 = 0.f;
#pragma unroll
    for (int kh = 0; kh < KS_; ++kh) {
      const float4* xr = (const float4*)(xb + (rr + kh) * XPITCH + c4);
      float4 xA = xr[0], xB = xr[1], xC = xr[2];   // LDS cols c4 .. c4+11
      float xv[12] = {xA.x, xA.y, xA.z, xA.w,
                      xB.x, xB.y, xB.z, xB.w,
                      xC.x, xC.y, xC.z, xC.w};
#pragma unroll
      for (int kw = 0; kw < KS_; ++kw) {
        const float4 wv =
            *(const float4*)&s_w[(kh * KS_ + kw) * (R_ * W_) + rr * W_ + c4];
        // pixel d needs global col (c4+d) + kw - 3  ==  LDS col c4 + d + kw + 1
        a0 = fmaf(xv[kw + 1], wv.x, a0);
        a1 = fmaf(xv[kw + 2], wv.y, a1);
        a2 = fmaf(xv[kw + 3], wv.z, a2);
        a3 = fmaf(xv[kw + 4], wv.w, a3);
      }
    }

    float* op = out + ((size_t)n * IC_ + (size_t)ch * WC_ + cw) * HW +
                (size_t)(h0 + rr) * W_ + c4;
    float4 o; o.x = a0; o.y = a1; o.z = a2; o.w = a3;
    *(float4*)op = o;         // coalesced b128 store

    __syncthreads();          // buf[ch&1] fully consumed before ch+2 overwrites it
  }
}

extern "C" void kernel_launch(void* const* d_in, const int* in_sizes, int n_in,
                              void* d_out, int out_size, void* d_ws, size_t ws_size,
                              hipStream_t stream) {
  (void)in_sizes; (void)n_in; (void)out_size; (void)d_ws; (void)ws_size;
  const float* x = (const float*)d_in[0];
  const float* w = (const float*)d_in[1];
  float* out = (float*)d_out;
  dim3 grid(H_ / R_, WC_, N_);   // 16 row tiles x 8 weight channels x 16 batch
  SKA_29583734735594_kernel<<<grid, TPB, 0, stream>>>(x, w, out);
}